// GraspObjectFeasibilityNet_37598143709461
// MI455X (gfx1250) — compile-verified
//
#include <hip/hip_runtime.h>
#include <hip/hip_bf16.h>
#include <stdint.h>

typedef __attribute__((ext_vector_type(2))) float v2f;
typedef __attribute__((ext_vector_type(8))) float v8f;

#define NPTS  2048
#define NSAMP 512
#define KNN   32
#define RAD2  0.04f

// ---------------------------------------------------------------------------
// WMMA helper: accumulate D += A(16xK) * B(KxN tile) using V_WMMA_F32_16X16X4_F32.
// aRow: pointer to A[mrow][0] (row stride == K, contiguous).
// wRow: pointer to W[nb+mrow][0] (row-major weights, row stride == K); B = W^T.
// Fragment layout per CDNA5 ISA 7.12.2:
//   A: lane(mrow,hi) holds A[mrow][kb+2*hi], A[mrow][kb+2*hi+1]
//   B: lane(mrow,hi) holds B[kb+2*hi][mrow], B[kb+2*hi+1][mrow] = W[n][k],W[n][k+1]
// ---------------------------------------------------------------------------
__device__ __forceinline__ v8f wmma_rowdot(const float* __restrict__ aRow,
                                           const float* __restrict__ wRow,
                                           int K, v8f acc, int hi)
{
  for (int kb = 0; kb < K; kb += 4) {
    v2f a, b;
    a.x = aRow[kb + 2 * hi];
    a.y = aRow[kb + 2 * hi + 1];
    b.x = wRow[kb + 2 * hi];
    b.y = wRow[kb + 2 * hi + 1];
    acc = __builtin_amdgcn_wmma_f32_16x16x4_f32(false, a, false, b, (short)0, acc,
                                                false, false);
  }
  return acc;
}

// ---------------------------------------------------------------------------
// Kernel 0: zero the objfeat accumulator (global atomic-max target).
// ---------------------------------------------------------------------------
__global__ void zero_kernel(float* __restrict__ p, int n)
{
  int i = blockIdx.x * 256 + threadIdx.x;
  if (i < n) p[i] = 0.f;
}

// ---------------------------------------------------------------------------
// Kernel 1: farthest point sampling. One block per batch, xyz+dist in LDS.
// Matches jnp.argmax first-occurrence tie-break (smaller index wins on ties).
// ---------------------------------------------------------------------------
__global__ __launch_bounds__(256)
void fps_kernel(const float* __restrict__ xyz, int* __restrict__ fidx,
                float* __restrict__ newxyz)
{
  __shared__ float sx[NPTS], sy[NPTS], sz[NPTS], sd[NPTS];
  __shared__ float rmax[256];
  __shared__ int   ridx[256];
  __shared__ int   sfar;

  int b = blockIdx.x;
  int t = threadIdx.x;
  const float* p = xyz + (long)b * NPTS * 3;

  for (int i = t; i < NPTS; i += 256) {
    sx[i] = p[i * 3 + 0];
    sy[i] = p[i * 3 + 1];
    sz[i] = p[i * 3 + 2];
    sd[i] = 1e10f;
  }
  if (t == 0) sfar = 0;
  __syncthreads();

  for (int s = 0; s < NSAMP; ++s) {
    int far = sfar;
    float cx = sx[far], cy = sy[far], cz = sz[far];
    if (t == 0) {
      fidx[b * NSAMP + s] = far;
      newxyz[((long)b * NSAMP + s) * 3 + 0] = cx;
      newxyz[((long)b * NSAMP + s) * 3 + 1] = cy;
      newxyz[((long)b * NSAMP + s) * 3 + 2] = cz;
    }
    float best = -1.f;
    int bidx = 0;
    for (int i = t; i < NPTS; i += 256) {
      float dx = sx[i] - cx, dy = sy[i] - cy, dz = sz[i] - cz;
      float d  = dx * dx + dy * dy + dz * dz;
      float nd = fminf(sd[i], d);
      sd[i] = nd;
      if (nd > best) { best = nd; bidx = i; }   // strict > keeps earliest index
    }
    rmax[t] = best;
    ridx[t] = bidx;
    __syncthreads();
    for (int off = 128; off > 0; off >>= 1) {
      if (t < off) {
        float ov = rmax[t + off];
        int   oi = ridx[t + off];
        if (ov > rmax[t] || (ov == rmax[t] && oi < ridx[t])) {
          rmax[t] = ov;
          ridx[t] = oi;
        }
      }
      __syncthreads();
    }
    if (t == 0) sfar = ridx[0];
    __syncthreads();
  }
}

// ---------------------------------------------------------------------------
// Kernel 2: ball query + grouping. One wave32 per (b,s) query.
// "sort then take first KNN within radius" == first KNN qualifying in scan
// order; pad with the first hit. Writes centered differences.
// ---------------------------------------------------------------------------
__global__ __launch_bounds__(256)
void ballquery_kernel(const float* __restrict__ xyz,
                      const float* __restrict__ newxyz,
                      float* __restrict__ grouped)
{
  int gwave = (blockIdx.x * 256 + threadIdx.x) >> 5;
  int lane  = threadIdx.x & 31;
  int b = gwave >> 9;          // /512
  int s = gwave & 511;

  const float* p = xyz + (long)b * NPTS * 3;
  long q = (long)b * NSAMP + s;
  float cx = newxyz[q * 3 + 0];
  float cy = newxyz[q * 3 + 1];
  float cz = newxyz[q * 3 + 2];
  float* g = grouped + q * KNN * 3;

  int count = 0;
  int firstIdx = -1;
  for (int base = 0; base < NPTS; base += 32) {
    if (count >= KNN) break;
    int i = base + lane;
    float dx = p[i * 3 + 0] - cx;
    float dy = p[i * 3 + 1] - cy;
    float dz = p[i * 3 + 2] - cz;
    float d = dx * dx + dy * dy + dz * dz;
    bool in = (d <= RAD2);
    unsigned mask = (unsigned)__ballot(in);
    int pre = __popc(mask & ((1u << lane) - 1u));
    int pos = count + pre;
    if (in && pos < KNN) {
      g[pos * 3 + 0] = dx;
      g[pos * 3 + 1] = dy;
      g[pos * 3 + 2] = dz;
    }
    if (firstIdx < 0 && mask != 0u) firstIdx = base + (__ffs((int)mask) - 1);
    count += __popc(mask);
  }
  if (count < KNN) {
    if (firstIdx < 0) firstIdx = NPTS - 1;   // degenerate clamp (can't happen)
    float fx = p[firstIdx * 3 + 0] - cx;
    float fy = p[firstIdx * 3 + 1] - cy;
    float fz = p[firstIdx * 3 + 2] - cz;
    for (int k = count + lane; k < KNN; k += 32) {
      g[k * 3 + 0] = fx;
      g[k * 3 + 1] = fy;
      g[k * 3 + 2] = fz;
    }
  }
}

// ---------------------------------------------------------------------------
// Kernel 3: SA1 grouped MLP 3->64->64->128 (+ReLU) and max over K=32.
// 8 waves/block; wave w owns rows of s-group (w>>1), k-half (w&1): 16 rows.
// Layer1 computed directly into C-layout; layers 2/3 via f32 WMMA with A
// fragments from a per-wave LDS slab and B fragments streamed from weights.
// ---------------------------------------------------------------------------
__global__ __launch_bounds__(256)
void sa1_mlp_kernel(const float* __restrict__ grouped,
                    const float* __restrict__ w0, const float* __restrict__ b0,  // 64x3
                    const float* __restrict__ w1, const float* __restrict__ b1,  // 64x64
                    const float* __restrict__ w2, const float* __restrict__ b2,  // 128x64
                    float* __restrict__ l1pts)
{
  __shared__ float slab[8][16][64];    // 32 KB, per-wave activation staging
  __shared__ float outmax[4][128];     //  2 KB, per-s max accumulator

  int t = threadIdx.x;
  int w = t >> 5, lane = t & 31;
  int mrow = lane & 15, hi = lane >> 4;
  int b = blockIdx.x >> 7;
  int sbase = (blockIdx.x & 127) * 4;
  int sg = w >> 1;                 // local s-group 0..3
  int s = sbase + sg;
  int khalf = w & 1;
  long Rbase = ((long)(b * NSAMP + s)) * KNN + 16 * khalf;

  for (int i = t; i < 4 * 128; i += 256) ((float*)outmax)[i] = 0.f;

  // this lane's 8 rows of centered xyz
  float gx[8], gy[8], gz[8];
  for (int r = 0; r < 8; ++r) {
    long row = Rbase + 8 * hi + r;
    gx[r] = grouped[row * 3 + 0];
    gy[r] = grouped[row * 3 + 1];
    gz[r] = grouped[row * 3 + 2];
  }
  __syncthreads();

  // layer 1: 3 -> 64, scalar in C-layout, ReLU, -> slab
  for (int tt = 0; tt < 4; ++tt) {
    int n = 16 * tt + mrow;
    float wx = w0[n * 3 + 0], wy = w0[n * 3 + 1], wz = w0[n * 3 + 2];
    float bb = b0[n];
    for (int r = 0; r < 8; ++r) {
      float v = fmaf(gx[r], wx, fmaf(gy[r], wy, fmaf(gz[r], wz, bb)));
      slab[w][8 * hi + r][n] = fmaxf(v, 0.f);
    }
  }
  __syncthreads();

  // layer 2: 64 -> 64 (WMMA), ReLU, kept in registers
  float h2[4][8];
  for (int tt = 0; tt < 4; ++tt) {
    v8f acc = {};
    int n = 16 * tt + mrow;
    acc = wmma_rowdot(&slab[w][mrow][0], &w1[n * 64], 64, acc, hi);
    float bb = b1[n];
    for (int r = 0; r < 8; ++r) h2[tt][r] = fmaxf(acc[r] + bb, 0.f);
  }
  __syncthreads();
  for (int tt = 0; tt < 4; ++tt)
    for (int r = 0; r < 8; ++r)
      slab[w][8 * hi + r][16 * tt + mrow] = h2[tt][r];
  __syncthreads();

  // layer 3: 64 -> 128 (WMMA), ReLU, max over rows -> LDS atomic max
  for (int tt = 0; tt < 8; ++tt) {
    v8f acc = {};
    int n = 16 * tt + mrow;
    acc = wmma_rowdot(&slab[w][mrow][0], &w2[n * 64], 64, acc, hi);
    float bb = b2[n];
    float m = 0.f;                       // post-ReLU values are >= 0
    for (int r = 0; r < 8; ++r) m = fmaxf(m, fmaxf(acc[r] + bb, 0.f));
    atomicMax((int*)&outmax[sg][n], __float_as_int(m));
  }
  __syncthreads();

  for (int i = t; i < 4 * 128; i += 256) {
    int sl = i >> 7, c = i & 127;
    l1pts[((long)(b * NSAMP + sbase + sl)) * 128 + c] = ((float*)outmax)[i];
  }
}

// ---------------------------------------------------------------------------
// Kernel 4: SA2 MLP 131->128->128->256 (+ReLU), max over 512 samples per batch
// via global int atomic-max. One wave32 per 16 rows.
// ---------------------------------------------------------------------------
__global__ __launch_bounds__(32)
void sa2_mlp_kernel(const float* __restrict__ newxyz, const float* __restrict__ l1pts,
                    const float* __restrict__ w0, const float* __restrict__ b0,  // 128x131
                    const float* __restrict__ w1, const float* __restrict__ b1,  // 128x128
                    const float* __restrict__ w2, const float* __restrict__ b2,  // 256x128
                    float* __restrict__ objfeat)
{
  __shared__ float feat[16][131];
  __shared__ float slab[16][128];

  int lane = threadIdx.x;
  int mrow = lane & 15, hi = lane >> 4;
  int b  = blockIdx.x >> 5;
  int j0 = (blockIdx.x & 31) * 16;

  for (int i = lane; i < 16 * 131; i += 32) {
    int r = i / 131, c = i % 131;
    long j = (long)b * NSAMP + j0 + r;
    feat[r][c] = (c < 3) ? newxyz[j * 3 + c] : l1pts[j * 128 + (c - 3)];
  }
  __syncthreads();

  // layer 1: 131 -> 128, scalar, ReLU -> slab
  for (int tt = 0; tt < 8; ++tt) {
    int n = 16 * tt + mrow;
    float acc[8];
    float bb = b0[n];
    for (int r = 0; r < 8; ++r) acc[r] = bb;
    const float* wr = &w0[n * 131];
    for (int c = 0; c < 131; ++c) {
      float wv = wr[c];
      for (int r = 0; r < 8; ++r) acc[r] = fmaf(feat[8 * hi + r][c], wv, acc[r]);
    }
    for (int r = 0; r < 8; ++r) slab[8 * hi + r][n] = fmaxf(acc[r], 0.f);
  }
  __syncthreads();

  // layer 2: 128 -> 128 (WMMA), ReLU, into registers
  float h2[8][8];
  for (int tt = 0; tt < 8; ++tt) {
    v8f acc = {};
    int n = 16 * tt + mrow;
    acc = wmma_rowdot(&slab[mrow][0], &w1[n * 128], 128, acc, hi);
    float bb = b1[n];
    for (int r = 0; r < 8; ++r) h2[tt][r] = fmaxf(acc[r] + bb, 0.f);
  }
  __syncthreads();
  for (int tt = 0; tt < 8; ++tt)
    for (int r = 0; r < 8; ++r)
      slab[8 * hi + r][16 * tt + mrow] = h2[tt][r];
  __syncthreads();

  // layer 3: 128 -> 256 (WMMA), ReLU, max over 16 rows -> global atomic max
  for (int tt = 0; tt < 16; ++tt) {
    v8f acc = {};
    int n = 16 * tt + mrow;
    acc = wmma_rowdot(&slab[mrow][0], &w2[n * 128], 128, acc, hi);
    float bb = b2[n];
    float m = 0.f;
    for (int r = 0; r < 8; ++r) m = fmaxf(m, fmaxf(acc[r] + bb, 0.f));
    atomicMax((int*)&objfeat[b * 256 + n], __float_as_int(m));
  }
}

// ---------------------------------------------------------------------------
// Kernel 5: pose encoder + fusion + heads. B=32 rows, single block, scalar.
// ---------------------------------------------------------------------------
__global__ __launch_bounds__(256)
void heads_kernel(const float* __restrict__ grasp, const float* __restrict__ initp,
                  const float* __restrict__ finalp, const int* __restrict__ surf,
                  const float* __restrict__ embI, const float* __restrict__ embF,
                  const float* __restrict__ pw0, const float* __restrict__ pb0,
                  const float* __restrict__ pw1, const float* __restrict__ pb1,
                  const float* __restrict__ pw2, const float* __restrict__ pb2,
                  const float* __restrict__ fw0, const float* __restrict__ fb0,
                  const float* __restrict__ fw1, const float* __restrict__ fb1,
                  const float* __restrict__ osw, const float* __restrict__ osb,
                  const float* __restrict__ ocw, const float* __restrict__ ocb,
                  const float* __restrict__ objfeat, float* __restrict__ out)
{
  __shared__ float x[32][37];
  __shared__ float hA[32][128];
  __shared__ float hB[32][128];
  int t = threadIdx.x;

  for (int i = t; i < 32 * 37; i += 256) {
    int row = i / 37, c = i % 37;
    float v;
    if (c < 7)       v = grasp[row * 7 + c];
    else if (c < 14) v = initp[row * 7 + (c - 7)];
    else if (c < 21) v = finalp[row * 7 + (c - 14)];
    else if (c < 29) v = embI[surf[row * 2 + 0] * 8 + (c - 21)];
    else             v = embF[surf[row * 2 + 1] * 8 + (c - 29)];
    x[row][c] = v;
  }
  __syncthreads();

  for (int i = t; i < 32 * 64; i += 256) {          // pe0: 37 -> 64
    int row = i >> 6, n = i & 63;
    float a = pb0[n];
    for (int c = 0; c < 37; ++c) a = fmaf(x[row][c], pw0[n * 37 + c], a);
    hA[row][n] = fmaxf(a, 0.f);
  }
  __syncthreads();
  for (int i = t; i < 32 * 128; i += 256) {         // pe1: 64 -> 128
    int row = i >> 7, n = i & 127;
    float a = pb1[n];
    for (int c = 0; c < 64; ++c) a = fmaf(hA[row][c], pw1[n * 64 + c], a);
    hB[row][n] = fmaxf(a, 0.f);
  }
  __syncthreads();
  for (int i = t; i < 32 * 128; i += 256) {         // pe2: 128 -> 128
    int row = i >> 7, n = i & 127;
    float a = pb2[n];
    for (int c = 0; c < 128; ++c) a = fmaf(hB[row][c], pw2[n * 128 + c], a);
    hA[row][n] = fmaxf(a, 0.f);
  }
  __syncthreads();
  for (int i = t; i < 32 * 128; i += 256) {         // fu0: [obj(256)|pose(128)] -> 128
    int row = i >> 7, n = i & 127;
    float a = fb0[n];
    const float* wr = &fw0[n * 384];
    const float* of = &objfeat[row * 256];
    for (int c = 0; c < 256; ++c) a = fmaf(of[c], wr[c], a);
    for (int c = 0; c < 128; ++c) a = fmaf(hA[row][c], wr[256 + c], a);
    hB[row][n] = fmaxf(a, 0.f);
  }
  __syncthreads();
  for (int i = t; i < 32 * 64; i += 256) {          // fu1: 128 -> 64
    int row = i >> 6, n = i & 63;
    float a = fb1[n];
    for (int c = 0; c < 128; ++c) a = fmaf(hB[row][c], fw1[n * 128 + c], a);
    hA[row][n] = fmaxf(a, 0.f);
  }
  __syncthreads();
  if (t < 64) {                                     // heads: 64 -> 1 each
    int row = t & 31;
    const float* wv = (t < 32) ? osw : ocw;
    float a = (t < 32) ? osb[0] : ocb[0];
    for (int c = 0; c < 64; ++c) a = fmaf(hA[row][c], wv[c], a);
    out[t] = a;   // out[0..31] = os, out[32..63] = oc
  }
}

// ---------------------------------------------------------------------------
extern "C" void kernel_launch(void* const* d_in, const int* in_sizes, int n_in,
                              void* d_out, int out_size, void* d_ws, size_t ws_size,
                              hipStream_t stream)
{
  const float* points = (const float*)d_in[0];
  const float* grasp  = (const float*)d_in[1];
  const float* initp  = (const float*)d_in[2];
  const float* finalp = (const float*)d_in[3];
  const int*   surf   = (const int*)d_in[4];
  const float* sa1_w0 = (const float*)d_in[5];
  const float* sa1_b0 = (const float*)d_in[6];
  const float* sa1_w1 = (const float*)d_in[7];
  const float* sa1_b1 = (const float*)d_in[8];
  const float* sa1_w2 = (const float*)d_in[9];
  const float* sa1_b2 = (const float*)d_in[10];
  const float* sa2_w0 = (const float*)d_in[11];
  const float* sa2_b0 = (const float*)d_in[12];
  const float* sa2_w1 = (const float*)d_in[13];
  const float* sa2_b1 = (const float*)d_in[14];
  const float* sa2_w2 = (const float*)d_in[15];
  const float* sa2_b2 = (const float*)d_in[16];
  const float* embI   = (const float*)d_in[17];
  const float* embF   = (const float*)d_in[18];
  const float* pw0 = (const float*)d_in[19]; const float* pb0 = (const float*)d_in[20];
  const float* pw1 = (const float*)d_in[21]; const float* pb1 = (const float*)d_in[22];
  const float* pw2 = (const float*)d_in[23]; const float* pb2 = (const float*)d_in[24];
  const float* fw0 = (const float*)d_in[25]; const float* fb0 = (const float*)d_in[26];
  const float* fw1 = (const float*)d_in[27]; const float* fb1 = (const float*)d_in[28];
  const float* osw = (const float*)d_in[29]; const float* osb = (const float*)d_in[30];
  const float* ocw = (const float*)d_in[31]; const float* ocb = (const float*)d_in[32];

  // workspace layout (bytes)
  char* ws = (char*)d_ws;
  int*   fidx    = (int*)(ws + 0);              //  32*512 int      = 64 KB
  float* newxyz  = (float*)(ws + 65536);        //  32*512*3 f32    = 192 KB
  float* grouped = (float*)(ws + 262144);       //  32*512*32*3 f32 = 6 MB
  float* l1pts   = (float*)(ws + 6553600);      //  32*512*128 f32  = 8 MB
  float* objfeat = (float*)(ws + 14942208);     //  32*256 f32      = 32 KB

  hipLaunchKernelGGL(zero_kernel, dim3(32), dim3(256), 0, stream, objfeat, 32 * 256);
  hipLaunchKernelGGL(fps_kernel, dim3(32), dim3(256), 0, stream, points, fidx, newxyz);
  hipLaunchKernelGGL(ballquery_kernel, dim3(2048), dim3(256), 0, stream,
                     points, newxyz, grouped);
  hipLaunchKernelGGL(sa1_mlp_kernel, dim3(4096), dim3(256), 0, stream,
                     grouped, sa1_w0, sa1_b0, sa1_w1, sa1_b1, sa1_w2, sa1_b2, l1pts);
  hipLaunchKernelGGL(sa2_mlp_kernel, dim3(1024), dim3(32), 0, stream,
                     newxyz, l1pts, sa2_w0, sa2_b0, sa2_w1, sa2_b1, sa2_w2, sa2_b2,
                     objfeat);
  hipLaunchKernelGGL(heads_kernel, dim3(1), dim3(256), 0, stream,
                     grasp, initp, finalp, surf, embI, embF,
                     pw0, pb0, pw1, pb1, pw2, pb2,
                     fw0, fb0, fw1, fb1, osw, osb, ocw, ocb,
                     objfeat, (float*)d_out);
}